// MemoryEfficientHMM_12541304504362
// MI455X (gfx1250) — compile-verified
//
#include <hip/hip_runtime.h>

#define B_ 16
#define S_ 1000
#define N_ 64
#define D_ 80
#define KAUG 160          // augmented K: [obs^2 (80) | obs (80)]
#define LOG2PI 1.8378770664093453f

typedef float v2f __attribute__((ext_vector_type(2)));
typedef float v8f __attribute__((ext_vector_type(8)));

// ---------------------------------------------------------------------------
// Kernel 0: one-time prep. Build augmented B matrix [N][KAUG] and the fused
// per-state constant  constAll[n] = -0.5*(sum logvar + D log2pi) - 0.5*sum m^2*iv
// ---------------------------------------------------------------------------
__global__ void hmm_prep(const float* __restrict__ means,
                         const float* __restrict__ logvars,
                         float* __restrict__ wsB,
                         float* __restrict__ wsC) {
    int n = threadIdx.x;
    if (n < N_) {
        float slog = 0.f, smm = 0.f;
        for (int d = 0; d < D_; ++d) {
            float lv = logvars[n * D_ + d];
            float iv = __expf(-lv);
            float m  = means[n * D_ + d];
            wsB[n * KAUG + d]       = iv;             // multiplies obs^2
            wsB[n * KAUG + D_ + d]  = -2.0f * m * iv; // multiplies obs
            slog += lv;
            smm  += m * m * iv;
        }
        wsC[n] = -0.5f * (slog + (float)D_ * LOG2PI) - 0.5f * smm;
    }
}

// ---------------------------------------------------------------------------
// Kernel 1: emission log-probs as one fused GEMM with fp32 WMMA.
// Grid: 1000 blocks x 128 threads. Block = 16 samples x 64 states.
// emis[(b*S+s)][n] = constAll[n] - 0.5 * sum_k A[s][k] * B[k][n]
// K-loop split into two statically-known regions -> branch-free unrolled body.
// ---------------------------------------------------------------------------
#define OBS_PAD 84   // 84 mod 64 = 20 -> conflict-free column reads
#define BPAD    164  // 164 mod 64 = 36 -> conflict-free; 656B rows stay 16B aligned

__global__ __launch_bounds__(128) void hmm_emis(
    const float* __restrict__ obs, const float* __restrict__ wsB,
    const float* __restrict__ wsC, float* __restrict__ emis) {
    __shared__ float s_obs[16 * OBS_PAD];
    __shared__ float s_B[N_ * BPAD];

    int tid  = threadIdx.x;
    int lane = tid & 31;
    int w    = tid >> 5;          // 4 waves = 4 N-tiles of 16 states
    int row0 = blockIdx.x * 16;

    // stage 16x80 obs tile
    for (int idx = tid; idx < 16 * D_; idx += 128) {
        int r = idx / D_, d = idx - r * D_;
        s_obs[r * OBS_PAD + d] = obs[(size_t)(row0 + r) * D_ + d];
    }
    // stage B matrix [64][160] (vectorized, rows stay 16B aligned in LDS)
    const float4* B4 = (const float4*)wsB;
    for (int idx = tid; idx < N_ * (KAUG / 4); idx += 128) {
        int n = idx / (KAUG / 4), c = idx - n * (KAUG / 4);
        *(float4*)&s_B[n * BPAD + c * 4] = B4[n * (KAUG / 4) + c];
    }
    __syncthreads();

    int n    = w * 16 + (lane & 15);
    int r    = lane & 15;
    int koff = (lane >> 4) << 1;   // ISA A/B frag: hi half-wave carries K+2

    const float* orow = &s_obs[r * OBS_PAD];
    const float* brow = &s_B[n * BPAD];

    v8f acc = {};
    // Region 1: k in [0,80)  -> A = obs^2
#pragma unroll
    for (int c = 0; c < D_ / 4; ++c) {
        int kk = c * 4 + koff;
        v2f a, b;
        float v0 = orow[kk];
        float v1 = orow[kk + 1];
        a.x = v0 * v0; a.y = v1 * v1;
        b.x = brow[kk];
        b.y = brow[kk + 1];
        acc = __builtin_amdgcn_wmma_f32_16x16x4_f32(
            false, a, false, b, (short)0, acc, false, false);
    }
    // Region 2: k in [80,160) -> A = obs
#pragma unroll
    for (int c = 0; c < D_ / 4; ++c) {
        int kk = c * 4 + koff;
        v2f a, b;
        a.x = orow[kk];
        a.y = orow[kk + 1];
        b.x = brow[D_ + kk];
        b.y = brow[D_ + kk + 1];
        acc = __builtin_amdgcn_wmma_f32_16x16x4_f32(
            false, a, false, b, (short)0, acc, false, false);
    }

    float cn   = wsC[n];
    int  rbase = (lane >> 4) << 3; // C layout: VGPR r -> row r (lo) / r+8 (hi)
#pragma unroll
    for (int r2 = 0; r2 < 8; ++r2) {
        int srow = row0 + rbase + r2;
        emis[(size_t)srow * N_ + n] = cn - 0.5f * acc[r2];
    }
}

// ---------------------------------------------------------------------------
// Kernel 2: sequential forward scan, single workgroup (128 threads, 4 waves).
// Per step: shifted-exp trick -> one 16x64x64 fp32-WMMA matvec in linear space.
// T fragments live in registers; alpha/p in padded LDS; next step's emission
// cachelines prefetched (global_prefetch_b8) to hide the serial dependence.
// ---------------------------------------------------------------------------
#define AP 68  // 68 mod 64 = 4 -> 16 distinct banks for fragment column reads

__global__ __launch_bounds__(128) void hmm_scan(
    const float* __restrict__ trans_logits,
    const float* __restrict__ init_logits,
    const float* __restrict__ emis,
    float* __restrict__ out) {
    __shared__ float s_T[N_ * AP];
    __shared__ float s_alpha[16 * AP];
    __shared__ float s_p[16 * AP];
    __shared__ float s_red[16 * 8];
    __shared__ float s_m[16];
    __shared__ float s_init[N_];
    __shared__ float s_misc[2];

    int tid  = threadIdx.x;
    int lane = tid & 31;
    int w    = tid >> 5;

    // T = exp(log_softmax(trans)) : one row per thread (one-time)
    if (tid < N_) {
        int i = tid;
        float mx = -3.4e38f;
        for (int j = 0; j < N_; ++j) mx = fmaxf(mx, trans_logits[i * N_ + j]);
        float s = 0.f;
        for (int j = 0; j < N_; ++j) s += __expf(trans_logits[i * N_ + j] - mx);
        float lse = __logf(s);
        for (int j = 0; j < N_; ++j)
            s_T[i * AP + j] = __expf(trans_logits[i * N_ + j] - mx - lse);
    }
    if (tid == 0) {
        float mx = -3.4e38f;
        for (int j = 0; j < N_; ++j) mx = fmaxf(mx, init_logits[j]);
        float s = 0.f;
        for (int j = 0; j < N_; ++j) s += __expf(init_logits[j] - mx);
        s_misc[0] = mx; s_misc[1] = __logf(s);
    }
    __syncthreads();
    if (tid < N_) s_init[tid] = init_logits[tid] - s_misc[0] - s_misc[1];
    __syncthreads();

    // alpha0 = init + emis[:,0]  (also write log_probs[:,0])
    for (int idx = tid; idx < B_ * N_; idx += 128) {
        int b = idx >> 6, j = idx & 63;
        float v = s_init[j] + emis[(size_t)(b * S_) * N_ + j];
        s_alpha[b * AP + j] = v;
        out[(size_t)(b * S_) * N_ + j] = v;
    }

    // hoist this wave's 16 B-fragments of T into registers (32 VGPRs)
    int n    = w * 16 + (lane & 15);
    int koff = (lane >> 4) << 1;
    v2f bf[16];
#pragma unroll
    for (int c = 0; c < 16; ++c) {
        int ki = c * 4 + koff;
        bf[c].x = s_T[ki * AP + n];
        bf[c].y = s_T[(ki + 1) * AP + n];
    }
    int rbase = (lane >> 4) << 3;
    int ba    = lane & 15;

    for (int t = 1; t < S_; ++t) {
        __syncthreads();
        // per-batch max of alpha: 8 partials/row then 16-thread finish
        {
            int b = tid >> 3, seg = tid & 7;
            const float* ar = &s_alpha[b * AP + seg * 8];
            float mx = ar[0];
#pragma unroll
            for (int q = 1; q < 8; ++q) mx = fmaxf(mx, ar[q]);
            s_red[b * 8 + seg] = mx;
        }
        __syncthreads();
        if (tid < 16) {
            float mx = s_red[tid * 8];
#pragma unroll
            for (int q = 1; q < 8; ++q) mx = fmaxf(mx, s_red[tid * 8 + q]);
            s_m[tid] = mx;
        }
        __syncthreads();
        // p = exp(alpha - m)
        for (int idx = tid; idx < B_ * N_; idx += 128) {
            int b = idx >> 6, j = idx & 63;
            s_p[b * AP + j] = __expf(s_alpha[b * AP + j] - s_m[b]);
        }
        __syncthreads();

        // load this step's emission values; prefetch next step's lines
        float e[8];
#pragma unroll
        for (int r2 = 0; r2 < 8; ++r2) {
            int b = rbase + r2;
            e[r2] = emis[((size_t)(b * S_ + t)) * N_ + n];
        }
        if (t + 1 < S_) {
#pragma unroll
            for (int r2 = 0; r2 < 8; ++r2) {
                int b = rbase + r2;
                __builtin_prefetch(&emis[((size_t)(b * S_ + t + 1)) * N_ + n], 0, 1);
            }
        }

        // 16x64 x 64x64 matvec: 16 fp32 WMMAs per wave
        v8f acc = {};
#pragma unroll
        for (int c = 0; c < 16; ++c) {
            int kk = c * 4 + koff;
            v2f a;
            a.x = s_p[ba * AP + kk];
            a.y = s_p[ba * AP + kk + 1];
            acc = __builtin_amdgcn_wmma_f32_16x16x4_f32(
                false, a, false, bf[c], (short)0, acc, false, false);
        }

        // alpha_t = m + log(acc) + e_t ; also stream out log_probs[:,t]
#pragma unroll
        for (int r2 = 0; r2 < 8; ++r2) {
            int b = rbase + r2;
            float v = s_m[b] + __logf(acc[r2]) + e[r2];
            s_alpha[b * AP + n] = v;
            out[((size_t)(b * S_ + t)) * N_ + n] = v;
        }
    }
    __syncthreads();

    // total_log_prob[b] = logsumexp_j alpha_T[b,:]
    if (tid < 16) {
        int b = tid;
        float mx = s_alpha[b * AP];
        for (int j = 1; j < N_; ++j) mx = fmaxf(mx, s_alpha[b * AP + j]);
        float s = 0.f;
        for (int j = 0; j < N_; ++j) s += __expf(s_alpha[b * AP + j] - mx);
        out[(size_t)B_ * S_ * N_ + b] = mx + __logf(s);
    }
}

// ---------------------------------------------------------------------------
extern "C" void kernel_launch(void* const* d_in, const int* in_sizes, int n_in,
                              void* d_out, int out_size, void* d_ws, size_t ws_size,
                              hipStream_t stream) {
    const float* obs   = (const float*)d_in[0]; // [B,S,D]
    const float* tl    = (const float*)d_in[1]; // [N,N]
    const float* means = (const float*)d_in[2]; // [N,D]
    const float* lvars = (const float*)d_in[3]; // [N,D]
    const float* il    = (const float*)d_in[4]; // [N]
    float* out = (float*)d_out;

    float* ws  = (float*)d_ws;
    float* wsB = ws;                      // [64][160]
    float* wsC = ws + N_ * KAUG;          // [64]
    float* wsE = ws + N_ * KAUG + N_;     // [B*S][64] emission log-probs

    hmm_prep<<<1, 64, 0, stream>>>(means, lvars, wsB, wsC);
    hmm_emis<<<(B_ * S_) / 16, 128, 0, stream>>>(obs, wsB, wsC, wsE);
    hmm_scan<<<1, 128, 0, stream>>>(tl, il, wsE, out);
}